// AdaptiveWaveletTransform_50912542326954
// MI455X (gfx1250) — compile-verified
//
#include <hip/hip_runtime.h>

// AdaptiveWaveletTransform for MI455X (gfx1250, wave32).
// B=64, T=8192, F=128, W=4, HID=64, 4 decimation levels.
//
// Pass 1: one sweep of the 256MB trajectory: WMMA f32 16x16x4 projection
//         (traj @ wavelet^T -> proj[B,T,4], scratch) + per-block column sums.
// Pass 2: tiny per-batch MLP + softmax -> weights[B,4] (deterministic sums).
// Pass 3: combined = proj . weights, scattered into 4 stride-decimated outputs.

typedef float v2f __attribute__((ext_vector_type(2)));
typedef float v8f __attribute__((ext_vector_type(8)));

constexpr int Bsz    = 64;
constexpr int Tdim   = 8192;
constexpr int Fdim   = 128;
constexpr int Wn     = 4;
constexpr int HID    = 64;
constexpr int PITCH  = 132;              // 128 + 4 dword pad -> conflict-free b64 LDS reads
constexpr int ROWS   = 64;               // t-rows per block (4 waves x 16)
constexpr int NCHUNK = Tdim / ROWS;      // 128 chunks per batch

// ---------------------------------------------------------------------------
// Kernel 1: per-chunk projection (WMMA) + per-chunk feature sums
// ---------------------------------------------------------------------------
__global__ __launch_bounds__(128)
void proj_partial_kernel(const float* __restrict__ traj,
                         const float* __restrict__ wav,
                         float* __restrict__ proj,       // [B, T, 4]
                         float* __restrict__ partial)    // [B, NCHUNK, F]
{
    __shared__ float tile[ROWS * PITCH];   // 64 x 132 f32  = 33792 B
    __shared__ float wlds[16 * PITCH];     // B-matrix, rows 4..15 zero = 8448 B
    __shared__ float part[4 * Fdim];       // per-warp-row column sums   = 2048 B

    const int tid   = threadIdx.x;
    const int b     = blockIdx.x / NCHUNK;
    const int chunk = blockIdx.x % NCHUNK;
    const int t0    = chunk * ROWS;

    // Zero B rows 4..15 (disjoint from rows 0..3 loaded below -> no barrier needed)
    for (int i = tid; i < 12 * PITCH; i += 128)
        wlds[4 * PITCH + i] = 0.0f;

    // Load wavelet_params [4,128] into B rows 0..3 (4 floats / thread)
    {
        const int i = tid * 4;                 // 0..508
        const int r = i >> 7, c = i & 127;
        float4 v = *(const float4*)(wav + r * Fdim + c);
        *(float4*)(&wlds[r * PITCH + c]) = v;
    }

    // Coalesced global->LDS copy of the 64x128 trajectory tile,
    // accumulating per-thread column sums from the in-flight registers.
    const float* src  = traj + ((size_t)b * Tdim + t0) * Fdim;
    const int col4  = tid & 31;                // float4 column 0..31
    const int rbase = tid >> 5;                // 0..3
    float4 a4 = make_float4(0.f, 0.f, 0.f, 0.f);
#pragma unroll
    for (int j = 0; j < 16; ++j) {
        const int row = rbase + 4 * j;
        float4 v = *(const float4*)(src + (size_t)row * Fdim + col4 * 4);
        *(float4*)(&tile[row * PITCH + col4 * 4]) = v;
        a4.x += v.x; a4.y += v.y; a4.z += v.z; a4.w += v.w;
    }
    *(float4*)(&part[rbase * Fdim + col4 * 4]) = a4;
    __syncthreads();

    // Deterministic per-chunk feature sums (fixed order, no atomics)
    {
        float s = part[tid] + part[Fdim + tid] + part[2 * Fdim + tid] + part[3 * Fdim + tid];
        partial[((size_t)b * NCHUNK + chunk) * Fdim + tid] = s;
    }

    // WMMA: wave w computes proj rows [t0+16w, t0+16w+16) for all 4 wavelets.
    // A fragment (16x4 f32): lane = (k>=2?16:0)+m ; VGPR0=K{0,2}, VGPR1=K{1,3}.
    const int wave = tid >> 5;
    const int lane = tid & 31;
    const int mn   = lane & 15;                // A-row m / B-col n
    const int half = lane >> 4;                // K half (0: K=0,1  1: K=2,3)
    const float* arow = &tile[(wave * 16 + mn) * PITCH];
    const float* brow = &wlds[mn * PITCH];

    v8f acc = {};
#pragma unroll
    for (int k0 = 0; k0 < Fdim; k0 += 4) {
        v2f a  = *(const v2f*)(arow + k0 + 2 * half);   // ds_load_b64, conflict-free
        v2f bb = *(const v2f*)(brow + k0 + 2 * half);
        acc = __builtin_amdgcn_wmma_f32_16x16x4_f32(
            /*neg_a=*/false, a, /*neg_b=*/false, bb,
            /*c_mod=*/(short)0, acc, /*reuse_a=*/false, /*reuse_b=*/false);
    }

    // D layout: lane(n=mn, half) VGPR v holds D[v + 8*half][n]
    if (mn < Wn) {
        float* dst = proj + ((size_t)b * Tdim + t0 + wave * 16 + 8 * half) * Wn + mn;
#pragma unroll
        for (int v = 0; v < 8; ++v)
            dst[v * Wn] = acc[v];
    }
}

// ---------------------------------------------------------------------------
// Kernel 2: per-batch selection MLP + softmax -> weights[B,4]
// ---------------------------------------------------------------------------
__global__ __launch_bounds__(64)
void weights_kernel(const float* __restrict__ partial,   // [B, NCHUNK, F]
                    const float* __restrict__ W1,        // [F, HID]
                    const float* __restrict__ b1,        // [HID]
                    const float* __restrict__ W2,        // [HID, W]
                    const float* __restrict__ b2,        // [W]
                    float* __restrict__ weights)         // [B, W]
{
    __shared__ float mean[Fdim];
    __shared__ float h[HID];
    __shared__ float logits[Wn];

    const int b = blockIdx.x, tid = threadIdx.x;

    // Deterministic fixed-order reduction over the 128 chunk partials
    const float* p = partial + (size_t)b * NCHUNK * Fdim;
    float s0 = 0.f, s1 = 0.f;
    for (int c = 0; c < NCHUNK; ++c) {
        s0 += p[(size_t)c * Fdim + tid];
        s1 += p[(size_t)c * Fdim + tid + 64];
    }
    const float invT = 1.0f / (float)Tdim;
    mean[tid]      = s0 * invT;
    mean[tid + 64] = s1 * invT;
    __syncthreads();

    // h = relu(mean @ W1 + b1)
    float acc = b1[tid];
#pragma unroll 4
    for (int f = 0; f < Fdim; ++f)
        acc += mean[f] * W1[f * HID + tid];
    h[tid] = fmaxf(acc, 0.0f);
    __syncthreads();

    // logits = h @ W2 + b2
    if (tid < Wn) {
        float l = b2[tid];
        for (int k = 0; k < HID; ++k)
            l += h[k] * W2[k * Wn + tid];
        logits[tid] = l;
    }
    __syncthreads();

    if (tid == 0) {
        float mx = fmaxf(fmaxf(logits[0], logits[1]), fmaxf(logits[2], logits[3]));
        float e0 = __expf(logits[0] - mx);
        float e1 = __expf(logits[1] - mx);
        float e2 = __expf(logits[2] - mx);
        float e3 = __expf(logits[3] - mx);
        float inv = 1.0f / (e0 + e1 + e2 + e3);
        weights[b * Wn + 0] = e0 * inv;
        weights[b * Wn + 1] = e1 * inv;
        weights[b * Wn + 2] = e2 * inv;
        weights[b * Wn + 3] = e3 * inv;
    }
}

// ---------------------------------------------------------------------------
// Kernel 3: combined = proj . weights ; scatter to 4 decimated outputs
// ---------------------------------------------------------------------------
__global__ __launch_bounds__(256)
void combine_kernel(const float* __restrict__ proj,     // [B, T, 4]
                    const float* __restrict__ weights,  // [B, 4]
                    float* __restrict__ out)            // concat of 4 levels
{
    const int tid = threadIdx.x;
    const int blocksPerBatch = Tdim / 256;              // 32
    const int b = blockIdx.x / blocksPerBatch;
    const int t = (blockIdx.x % blocksPerBatch) * 256 + tid;

    const float4 w = *(const float4*)(weights + b * Wn);
    const float4 p = *(const float4*)(proj + ((size_t)b * Tdim + t) * Wn);
    const float c = p.x * w.x + p.y * w.y + p.z * w.z + p.w * w.w;

    float* out0 = out;                                  // [64, 8192]
    float* out1 = out0 + (size_t)Bsz * Tdim;            // [64, 4096]
    float* out2 = out1 + (size_t)Bsz * (Tdim / 2);      // [64, 2048]
    float* out3 = out2 + (size_t)Bsz * (Tdim / 4);      // [64, 1024]

    out0[(size_t)b * Tdim + t] = c;
    if ((t & 1) == 0) out1[(size_t)b * (Tdim / 2) + (t >> 1)] = c;
    if ((t & 3) == 0) out2[(size_t)b * (Tdim / 4) + (t >> 2)] = c;
    if ((t & 7) == 0) out3[(size_t)b * (Tdim / 8) + (t >> 3)] = c;
}

// ---------------------------------------------------------------------------
extern "C" void kernel_launch(void* const* d_in, const int* in_sizes, int n_in,
                              void* d_out, int out_size, void* d_ws, size_t ws_size,
                              hipStream_t stream)
{
    (void)in_sizes; (void)n_in; (void)out_size; (void)ws_size;

    const float* traj = (const float*)d_in[0];   // [64, 8192, 128]
    const float* wav  = (const float*)d_in[1];   // [4, 128]
    const float* W1   = (const float*)d_in[2];   // [128, 64]
    const float* b1   = (const float*)d_in[3];   // [64]
    const float* W2   = (const float*)d_in[4];   // [64, 4]
    const float* b2   = (const float*)d_in[5];   // [4]
    float* out = (float*)d_out;

    // Workspace layout: proj (8 MiB) | partial (4 MiB) | weights (1 KiB)
    char* ws = (char*)d_ws;
    const size_t projBytes    = (size_t)Bsz * Tdim * Wn * sizeof(float);
    const size_t partialBytes = (size_t)Bsz * NCHUNK * Fdim * sizeof(float);
    float* proj    = (float*)ws;
    float* partial = (float*)(ws + projBytes);
    float* weights = (float*)(ws + projBytes + partialBytes);

    proj_partial_kernel<<<Bsz * NCHUNK, 128, 0, stream>>>(traj, wav, proj, partial);
    weights_kernel<<<Bsz, 64, 0, stream>>>(partial, W1, b1, W2, b2, weights);
    combine_kernel<<<Bsz * (Tdim / 256), 256, 0, stream>>>(proj, weights, out);
}